// Loss_NeuroMUSX_77068893159481
// MI455X (gfx1250) — compile-verified
//
#include <hip/hip_runtime.h>

// ---------------------------------------------------------------------------
// Segmented BCE loss for MI455X (gfx1250, wave32).
//
// Roofline: must stream pred[N]+targ[N]+mask[N] = 96 MB (batch is sorted, so
// segment bounds come from 2048 binary searches, not a 32 MB stream). At
// 23.3 TB/s that is ~4.2 us. To actually sit on that roofline the per-element
// math must be thin: hardware v_log_f32 (1 TRANS op, co-executing with VALU)
// replaces the ~40-op precise logf/log1pf expansions. Loads are NT b128
// (read-once data, keep the 192 MB L2 clean, 4x fewer VMEM instructions).
// Finalize: a single wave32 reduces 2x2048 f32 values with chained
// v_wmma_f32_16x16x4_f32 (ones-matrix row-sum trick, 64 values per WMMA).
// All reduction orders fixed -> deterministic output, no atomics.
// ---------------------------------------------------------------------------

typedef __attribute__((ext_vector_type(2))) float v2f;
typedef __attribute__((ext_vector_type(4))) float v4f;
typedef __attribute__((ext_vector_type(8))) float v8f;
typedef __attribute__((ext_vector_type(4))) int   v4i;

#define LN2F 0.69314718055994530942f

// BCE with jnp.clip(log, -100) semantics via hardware log2:
// v_log_f32(0) = -inf -> fmax clamps to -100, matching the reference.
__device__ __forceinline__ float bce_fast(float p, float t) {
    float lp = fmaxf(__log2f(p) * LN2F,        -100.0f);
    float lq = fmaxf(__log2f(1.0f - p) * LN2F, -100.0f);
    // t*lp + (1-t)*lq = lq + t*(lp - lq)
    return -(lq + t * (lp - lq));
}

__device__ __forceinline__ float wave_reduce_sum(float v) {
    #pragma unroll
    for (int off = 16; off > 0; off >>= 1)
        v += __shfl_xor(v, off, 32);
    return v;
}

// One block per segment (batch sorted). Streams [start,end) with NT b128
// loads; scalar head/tail handle the 16B alignment of `start`.
__global__ void __launch_bounds__(256)
seg_bce_kernel(const float* __restrict__ pred,
               const float* __restrict__ targ,
               const int*   __restrict__ batch,
               const int*   __restrict__ mask,
               float*       __restrict__ per_graph,
               int N) {
    const int b   = (int)blockIdx.x;
    const int tid = (int)threadIdx.x;

    __shared__ int s_range[2];
    if (tid == 0) {
        int lo = 0, hi = N;                       // lower_bound(batch, b)
        while (lo < hi) { int mid = (lo + hi) >> 1; if (batch[mid] < b) lo = mid + 1; else hi = mid; }
        const int start = lo;
        hi = N;                                   // lower_bound(batch, b+1)
        while (lo < hi) { int mid = (lo + hi) >> 1; if (batch[mid] < b + 1) lo = mid + 1; else hi = mid; }
        s_range[0] = start;
        s_range[1] = lo;
    }
    __syncthreads();
    const int start = s_range[0];
    const int end   = s_range[1];

    float bce_acc = 0.0f;
    float cnt_acc = 0.0f;

    // head: up to 3 scalar elements to reach 16B alignment
    int a0 = (start + 3) & ~3;
    if (a0 > end) a0 = end;
    if (tid < (a0 - start)) {
        const int i = start + tid;
        float m = (float)mask[i];
        bce_acc += bce_fast(pred[i] * m, targ[i]);
        cnt_acc += m;
    }

    // vectorized body: NT b128 loads, coalesced across the wave
    const int  nvec  = (end - a0) >> 2;
    const v4f* pred4 = (const v4f*)(pred + a0);
    const v4f* targ4 = (const v4f*)(targ + a0);
    const v4i* mask4 = (const v4i*)(mask + a0);
    #pragma unroll 2
    for (int v = tid; v < nvec; v += 256) {
        v4f p = __builtin_nontemporal_load(&pred4[v]);
        v4f t = __builtin_nontemporal_load(&targ4[v]);
        v4i q = __builtin_nontemporal_load(&mask4[v]);
        #pragma unroll
        for (int k = 0; k < 4; ++k) {
            float m = (float)q[k];
            bce_acc += bce_fast(p[k] * m, t[k]);   // masked nodes still contribute
            cnt_acc += m;
        }
    }

    // tail: up to 3 scalar elements
    for (int i = a0 + (nvec << 2) + tid; i < end; i += 256) {
        float m = (float)mask[i];
        bce_acc += bce_fast(pred[i] * m, targ[i]);
        cnt_acc += m;
    }

    bce_acc = wave_reduce_sum(bce_acc);
    cnt_acc = wave_reduce_sum(cnt_acc);

    __shared__ float s_b[8];
    __shared__ float s_c[8];
    const int lane = tid & 31, wid = tid >> 5;
    if (lane == 0) { s_b[wid] = bce_acc; s_c[wid] = cnt_acc; }
    __syncthreads();

    if (tid == 0) {
        float tb = 0.0f, tc = 0.0f;
        #pragma unroll
        for (int w = 0; w < 8; ++w) { tb += s_b[w]; tc += s_c[w]; }
        per_graph[b] = (tc > 0.0f) ? (tb / fmaxf(tc, 1.0f)) : 0.0f;
    }
}

// Single wave32. Reduces per_graph[B] plus the (L1/B)-scaled saturation BCE
// via chained V_WMMA_F32_16X16X4_F32 with an all-ones B operand:
//   D[m][n] = sum_k A[m][k] + C[m][n]   -> 64 f32 folded per WMMA.
__global__ void __launch_bounds__(32)
finalize_wmma_kernel(const float* __restrict__ per_graph,
                     const float* __restrict__ sat_pred,
                     const float* __restrict__ sat_targ,
                     float*       __restrict__ out,
                     int B) {
    const int lane = (int)threadIdx.x;                  // 0..31, EXEC all ones
    const float sat_scale = (1.0f / 50.0f) / (float)B;  // L1 * mean

    v2f ones; ones.x = 1.0f; ones.y = 1.0f;
    v8f acc = {};

    const int Bv = B & ~63;                             // WMMA chunks of 64
    for (int c = 0; c < Bv; c += 64) {
        v2f a;
        a.x = per_graph[c + lane];                      // any bijection keeps the total
        a.y = per_graph[c + 32 + lane];
        acc = __builtin_amdgcn_wmma_f32_16x16x4_f32(
                  false, a, false, ones, (short)0, acc, false, false);
        const int i0 = c + lane, i1 = c + 32 + lane;
        a.x = bce_fast(sat_pred[i0], sat_targ[i0]) * sat_scale;
        a.y = bce_fast(sat_pred[i1], sat_targ[i1]) * sat_scale;
        acc = __builtin_amdgcn_wmma_f32_16x16x4_f32(
                  false, a, false, ones, (short)0, acc, false, false);
    }

    // scalar tail (dead for B=2048, kept for safety)
    float tail = 0.0f;
    for (int i = Bv + lane; i < B; i += 32) {
        tail += per_graph[i] + bce_fast(sat_pred[i], sat_targ[i]) * sat_scale;
    }
    tail = wave_reduce_sum(tail);

    // collapse tile: all columns identical; rows 0..7 = vgpr0..7 @ lane 0,
    // rows 8..15 = vgpr0..7 @ lane 16.
    float s = acc[0] + acc[1] + acc[2] + acc[3] + acc[4] + acc[5] + acc[6] + acc[7];
    float total = __shfl(s, 0, 32) + __shfl(s, 16, 32) + tail;

    if (lane == 0) out[0] = total;
}

extern "C" void kernel_launch(void* const* d_in, const int* in_sizes, int n_in,
                              void* d_out, int out_size, void* d_ws, size_t ws_size,
                              hipStream_t stream) {
    const float* y_mus_pred = (const float*)d_in[0];   // [N] f32
    const float* y_mus      = (const float*)d_in[1];   // [N] f32
    const float* y_sat_pred = (const float*)d_in[2];   // [B] f32
    const float* y_sat      = (const float*)d_in[3];   // [B] f32
    const int*   batch      = (const int*)  d_in[4];   // [N] i32 sorted
    const int*   mask       = (const int*)  d_in[5];   // [N] i32 {0,1}

    const int N = in_sizes[0];
    const int B = in_sizes[2];

    float* per_graph = (float*)d_ws;                   // B floats of scratch
    float* out       = (float*)d_out;                  // scalar f32

    seg_bce_kernel<<<B, 256, 0, stream>>>(y_mus_pred, y_mus, batch, mask,
                                          per_graph, N);
    finalize_wmma_kernel<<<1, 32, 0, stream>>>(per_graph, y_sat_pred, y_sat,
                                               out, B);
}